// GNNModel_11278584119897
// MI455X (gfx1250) — compile-verified
//
#include <hip/hip_runtime.h>
#include <hip/hip_bf16.h>

typedef __attribute__((ext_vector_type(2))) float v2f;
typedef __attribute__((ext_vector_type(4))) float v4f;
typedef __attribute__((ext_vector_type(8))) float v8f;

#define NODES 100000
#define EDGES 3200000
#define NODE_F 128
#define EDGE_F 16
#define OUT_C 15

// ---------------------------------------------------------------------------
// C[M,15] = A[M,K] @ W[K,15]  via V_WMMA_F32_16X16X4_F32.
// One wave32 computes one 16x16 output tile (col 15 is zero padding).
//  - W staged in LDS, padded to 16 cols, interleaved [k/2][col][k&1] so each
//    lane fetches its (W[kb][r], W[kb+1][r]) pair with one ds_load_b64.
//  - Output tile transposed through LDS so the 240 contiguous floats at
//    C + m0*15 are written with coalesced b128 stores.
// M is an exact multiple of 16 at both call sites (100000, 3200000).
// ---------------------------------------------------------------------------
template <int K>
__global__ __launch_bounds__(256) void gemm_w15_wmma(
    const float* __restrict__ A, const float* __restrict__ W,
    float* __restrict__ C, int M) {
  __shared__ float sW[K * 16];        // padded+interleaved weights
  __shared__ float sO[8 * 240];       // per-wave output staging (8 waves)

  // Cooperative staged load of W: element W[k][c] -> sW[(k>>1)*32 + c*2 + (k&1)]
  // (uniform trip count: K*16 is a multiple of blockDim)
  for (int i = threadIdx.x; i < K * 16; i += 256) {
    const int k = i >> 4, c = i & 15;
    sW[(k >> 1) * 32 + c * 2 + (k & 1)] = (c < 15) ? W[k * OUT_C + c] : 0.0f;
  }
  __syncthreads();

  const int lane = threadIdx.x & 31;
  const int wave = threadIdx.x >> 5;
  const long long tile = (long long)blockIdx.x * (blockDim.x >> 5) + wave;
  const long long m0 = tile * 16;
  if (m0 >= M) return;                        // wave-uniform exit

  const int r  = lane & 15;                   // A row in tile / B-D column
  const int kh = (lane >> 4) * 2;             // K sub-offset: 0 or 2
  const int hi = lane >> 4;                   // half-wave select
  const float* __restrict__ arow = A + (m0 + r) * (long long)K;
  const float* sWl = sW + r * 2 + hi * 32;    // lane's B base (kb even)

  v8f acc = {};
#pragma unroll
  for (int k0 = 0; k0 < K; k0 += 4) {
    v2f a;
    a.x = arow[k0 + kh];                      // global_load_b64
    a.y = arow[k0 + kh + 1];
    const v2f b = *(const v2f*)(sWl + k0 * 16);   // ds_load_b64, no branches
    acc = __builtin_amdgcn_wmma_f32_16x16x4_f32(
        /*neg_a=*/false, a, /*neg_b=*/false, b,
        /*c_mod=*/(short)0, acc, /*reuse_a=*/false, /*reuse_b=*/false);
  }

  // Transpose tile through LDS, then write 240 contiguous floats coalesced.
  float* so = sO + wave * 240;
  if (r < 15) {
#pragma unroll
    for (int v = 0; v < 8; ++v)
      so[(v + (hi << 3)) * OUT_C + r] = acc[v];   // ds_store_b32
  }
  __builtin_amdgcn_wave_barrier();   // DS ops of a wave execute in order
  float* __restrict__ crow = C + m0 * OUT_C;      // 16B-aligned (tile*960 B)
#pragma unroll
  for (int it = 0; it < 2; ++it) {
    const int idx = lane + it * 32;
    if (idx < 60)
      *(v4f*)(crow + idx * 4) = *(const v4f*)(so + idx * 4);  // b128
  }
}

__global__ void zero_f32(float* __restrict__ p, long long n) {
  long long i = (long long)blockIdx.x * blockDim.x + threadIdx.x;
  if (i < n) p[i] = 0.0f;
}

// axw[row] += adj[e] * xw[col]  (15 contiguous f32 atomics per edge)
__global__ void edge_scatter(const long long* __restrict__ eidx,
                             const float* __restrict__ adj,
                             const float* __restrict__ xw,
                             float* __restrict__ axw, int E) {
  int e = blockIdx.x * blockDim.x + threadIdx.x;
  if (e >= E) return;
  const long long row = eidx[e];
  const long long col = eidx[(long long)E + e];
  const float v = adj[e];
  const float* __restrict__ s = xw + col * OUT_C;
  float* __restrict__ d = axw + row * OUT_C;
#pragma unroll
  for (int j = 0; j < OUT_C; ++j)
    atomicAdd(d + j, v * s[j]);              // global_atomic_add_f32, no return
}

// scores[e] = sigmoid( sum_c axw[row][c] * ew[e][c] * axw[col][c] )
__global__ void edge_score(const long long* __restrict__ eidx,
                           const float* __restrict__ ew,
                           const float* __restrict__ axw,
                           float* __restrict__ scores, int E) {
  int e = blockIdx.x * blockDim.x + threadIdx.x;
  if (e >= E) return;
  const long long row = eidx[e];
  const long long col = eidx[(long long)E + e];
  const float* __restrict__ h = axw + row * OUT_C;
  const float* __restrict__ t = axw + col * OUT_C;
  const float* __restrict__ w = ew + (long long)e * OUT_C;
  float s = 0.0f;
#pragma unroll
  for (int j = 0; j < OUT_C; ++j)
    s = fmaf(h[j] * w[j], t[j], s);
  scores[e] = 1.0f / (1.0f + __expf(-s));
}

extern "C" void kernel_launch(void* const* d_in, const int* in_sizes, int n_in,
                              void* d_out, int out_size, void* d_ws, size_t ws_size,
                              hipStream_t stream) {
  const float*     x   = (const float*)d_in[0];      // [N,128]
  const long long* ei  = (const long long*)d_in[1];  // [2,E] int64
  const float*     ea  = (const float*)d_in[2];      // [E,16]
  const float*     wn  = (const float*)d_in[3];      // [128,15]
  const float*     we  = (const float*)d_in[4];      // [16,15]
  const float*     adj = (const float*)d_in[5];      // [E]

  float* axw = (float*)d_out;                            // [N,15]
  float* ew  = axw + (long long)NODES * OUT_C;           // [E,15]
  float* sc  = ew + (long long)EDGES * OUT_C;            // [E]

  // xw = x @ W_node scratch (6 MB). Fall back to aliasing the scores region
  // (12.8 MB, written only by the final kernel) if the workspace is small.
  const size_t xw_bytes = (size_t)NODES * OUT_C * sizeof(float);
  float* xw = (ws_size >= xw_bytes) ? (float*)d_ws : sc;

  // 1) xw = x @ W_node   (WMMA f32, K=128)
  {
    const int tiles  = NODES / 16;                 // 6250
    const int blocks = (tiles + 7) / 8;
    gemm_w15_wmma<NODE_F><<<blocks, 256, 0, stream>>>(x, wn, xw, NODES);
  }
  // 2) ew = edge_attr @ W_edge   (WMMA f32, K=16)
  {
    const int tiles  = EDGES / 16;                 // 200000
    const int blocks = (tiles + 7) / 8;
    gemm_w15_wmma<EDGE_F><<<blocks, 256, 0, stream>>>(ea, we, ew, EDGES);
  }
  // 3) axw = 0
  {
    const long long n = (long long)NODES * OUT_C;
    zero_f32<<<(int)((n + 255) / 256), 256, 0, stream>>>(axw, n);
  }
  // 4) axw[row] += adj * xw[col]   (segment-sum refactored through W_node)
  edge_scatter<<<(EDGES + 255) / 256, 256, 0, stream>>>(ei, adj, xw, axw, EDGES);
  // 5) scores
  edge_score<<<(EDGES + 255) / 256, 256, 0, stream>>>(ei, ew, axw, sc, EDGES);
}